// Aggregator_46205258170763
// MI455X (gfx1250) — compile-verified
//
#include <hip/hip_runtime.h>

// ---------------------------------------------------------------------------
// KGAT-style aggregator for MI455X (gfx1250, wave32).
// Phases:
//   1) KG edge scatter-mean      (atomic f32, bandwidth bound)
//   2) interaction scatter-mean  (atomic f32, bandwidth bound)
//   3) mean-finalize + gate GEMMs (v_wmma_f32_16x16x4_f32) + sigmoid fusion
//   4) attribute-entity mean finalize
//   5) user scatter-sum of fused item embeddings
// ---------------------------------------------------------------------------

#define DD 64  // embedding dim

typedef __attribute__((ext_vector_type(2))) float v2f;
typedef __attribute__((ext_vector_type(8))) float v8f;

// ---- 1) KG edge scatter: ent_sum[head] += entity_emb[tail]*weight[rel] ----
// slot = (edge, pair); pair in [0,32) covers dims {2*pair, 2*pair+1}
__global__ void kg_edge_scatter(const float* __restrict__ entity_emb,
                                const int* __restrict__ edge_head,
                                const int* __restrict__ edge_tail,
                                const int* __restrict__ edge_type,
                                const float* __restrict__ weight,
                                float* __restrict__ ent_sum,
                                float* __restrict__ ent_cnt,
                                int n_edges)
{
    long long t = (long long)blockIdx.x * blockDim.x + threadIdx.x;
    long long total = (long long)n_edges * 32;
    if (t >= total) return;
    int e    = (int)(t >> 5);
    int pair = (int)(t & 31);
    int head = edge_head[e];
    int tail = edge_tail[e];
    int rel  = edge_type[e];
    float2 v = ((const float2*)(entity_emb + (long long)tail * DD))[pair];
    float2 w = ((const float2*)(weight     + (long long)rel  * DD))[pair];
    float* dst = ent_sum + (long long)head * DD + pair * 2;
    atomicAdd(dst + 0, v.x * w.x);
    atomicAdd(dst + 1, v.y * w.y);
    if (pair == 0) atomicAdd(ent_cnt + head, 1.0f);
}

// ---- 2) interaction scatter: int_sum[col] += user_emb[row] ----
__global__ void inter_scatter(const float* __restrict__ user_emb,
                              const int* __restrict__ rows,
                              const int* __restrict__ cols,
                              float* __restrict__ sum,
                              float* __restrict__ cnt,
                              int n)
{
    long long t = (long long)blockIdx.x * blockDim.x + threadIdx.x;
    long long total = (long long)n * 32;
    if (t >= total) return;
    int i    = (int)(t >> 5);
    int pair = (int)(t & 31);
    int r = rows[i];
    int c = cols[i];
    float2 v = ((const float2*)(user_emb + (long long)r * DD))[pair];
    float* dst = sum + (long long)c * DD + pair * 2;
    atomicAdd(dst + 0, v.x);
    atomicAdd(dst + 1, v.y);
    if (pair == 0) atomicAdd(cnt + c, 1.0f);
}

// ---- 3) mean finalize + WMMA gate GEMMs + sigmoid fusion ----
// Block: 128 threads (4 waves); block owns 64 item rows; wave owns 16 rows.
// gi = sigmoid(kg @ G1^T + it @ G2^T); fe = gi*kg + (1-gi)*it.
__global__ void gate_fuse_wmma(const float* __restrict__ ent_sum,
                               const float* __restrict__ ent_cnt,
                               const float* __restrict__ int_sum,
                               const float* __restrict__ int_cnt,
                               const float* __restrict__ g1,
                               const float* __restrict__ g2,
                               float* __restrict__ out_fe,
                               float* __restrict__ out_kg,
                               float* __restrict__ out_in,
                               int n_items)
{
    // stride 65 => bank-conflict-free column reads on 64-bank LDS
    __shared__ float s_kg[64 * 65];
    __shared__ float s_it[64 * 65];
    __shared__ float s_g1[64 * 65];
    __shared__ float s_g2[64 * 65];

    const int tid = threadIdx.x;
    const int rowBase = blockIdx.x * 64;

    // Stage means (and gate weights) into LDS; emit item_kg / item_int outputs.
    for (int idx = tid; idx < 64 * 64; idx += blockDim.x) {
        int r = idx >> 6, c = idx & 63;
        int grow = rowBase + r;
        int lrow = (grow < n_items) ? grow : (n_items - 1);  // clamp loads
        float kc = ent_cnt[lrow]; kc = kc > 1.0f ? kc : 1.0f;
        float ic = int_cnt[lrow]; ic = ic > 1.0f ? ic : 1.0f;
        float kv = ent_sum[(long long)lrow * DD + c] / kc;
        float iv = int_sum[(long long)lrow * DD + c] / ic;
        s_kg[r * 65 + c] = kv;
        s_it[r * 65 + c] = iv;
        s_g1[r * 65 + c] = g1[idx];
        s_g2[r * 65 + c] = g2[idx];
        if (grow < n_items) {
            out_kg[(long long)grow * DD + c] = kv;
            out_in[(long long)grow * DD + c] = iv;
        }
    }
    __syncthreads();

    const int lane     = tid & 31;
    const int wave     = tid >> 5;      // 0..3 -> which 16-row M tile
    const int mBase    = wave * 16;
    const int laneHalf = lane >> 4;     // 0|1: A holds K={0,1} vs K={2,3}
    const int lane16   = lane & 15;
    const int arow     = mBase + lane16;

    for (int nt = 0; nt < 4; ++nt) {
        const int colBase = nt * 16;
        const int bcol = colBase + lane16;
        v8f acc = {};
        // kg @ G1^T : B[k][n] = g1[n][k]
        #pragma unroll
        for (int ks = 0; ks < 16; ++ks) {
            int kb = ks * 4 + laneHalf * 2;
            v2f a, b;
            a.x = s_kg[arow * 65 + kb];  a.y = s_kg[arow * 65 + kb + 1];
            b.x = s_g1[bcol * 65 + kb];  b.y = s_g1[bcol * 65 + kb + 1];
            acc = __builtin_amdgcn_wmma_f32_16x16x4_f32(false, a, false, b,
                                                        (short)0, acc, false, false);
        }
        // + it @ G2^T
        #pragma unroll
        for (int ks = 0; ks < 16; ++ks) {
            int kb = ks * 4 + laneHalf * 2;
            v2f a, b;
            a.x = s_it[arow * 65 + kb];  a.y = s_it[arow * 65 + kb + 1];
            b.x = s_g2[bcol * 65 + kb];  b.y = s_g2[bcol * 65 + kb + 1];
            acc = __builtin_amdgcn_wmma_f32_16x16x4_f32(false, a, false, b,
                                                        (short)0, acc, false, false);
        }
        // D layout: VGPR v -> row mBase + v + 8*laneHalf, col = colBase + lane16
        #pragma unroll
        for (int v = 0; v < 8; ++v) {
            int m = mBase + v + 8 * laneHalf;
            int n = colBase + lane16;
            float x  = acc[v];
            float gi = 1.0f / (1.0f + __expf(-x));
            float kv = s_kg[m * 65 + n];
            float iv = s_it[m * 65 + n];
            float fu = gi * kv + (1.0f - gi) * iv;
            int grow = rowBase + m;
            if (grow < n_items) out_fe[(long long)grow * DD + n] = fu;
        }
    }
}

// ---- 4) attribute-entity mean finalize: rows [n_items, n_entities) ----
__global__ void att_mean(const float* __restrict__ ent_sum,
                         const float* __restrict__ ent_cnt,
                         float* __restrict__ out_fe,
                         int n_items, int n_entities)
{
    long long t = (long long)blockIdx.x * blockDim.x + threadIdx.x;
    long long total = (long long)(n_entities - n_items) * DD;
    if (t >= total) return;
    long long row = n_items + (t >> 6);
    int c = (int)(t & 63);
    float cn = ent_cnt[row]; cn = cn > 1.0f ? cn : 1.0f;
    out_fe[row * DD + c] = ent_sum[row * DD + c] / cn;
}

// ---- 5) user scatter-sum of fused item rows ----
__global__ void user_agg_scatter(const float* __restrict__ fusion,  // out_fe rows < n_items
                                 const int* __restrict__ rows,
                                 const int* __restrict__ cols,
                                 float* __restrict__ out_ua,
                                 int n)
{
    long long t = (long long)blockIdx.x * blockDim.x + threadIdx.x;
    long long total = (long long)n * 32;
    if (t >= total) return;
    int i    = (int)(t >> 5);
    int pair = (int)(t & 31);
    int r = rows[i];
    int c = cols[i];
    float2 v = ((const float2*)(fusion + (long long)c * DD))[pair];
    float* dst = out_ua + (long long)r * DD + pair * 2;
    atomicAdd(dst + 0, v.x);
    atomicAdd(dst + 1, v.y);
}

extern "C" void kernel_launch(void* const* d_in, const int* in_sizes, int n_in,
                              void* d_out, int out_size, void* d_ws, size_t ws_size,
                              hipStream_t stream)
{
    constexpr int N_ENT = 300000, N_IT = 150000, N_US = 150000;

    const float* entity_emb = (const float*)d_in[0];
    const float* user_emb   = (const float*)d_in[1];
    const int*   edge_index = (const int*)d_in[2];   // (2, E) flat
    const int*   edge_type  = (const int*)d_in[3];
    const int*   mat_row    = (const int*)d_in[4];
    const int*   mat_col    = (const int*)d_in[5];
    const float* weight     = (const float*)d_in[6];
    const float* g1         = (const float*)d_in[7];
    const float* g2         = (const float*)d_in[8];
    const int n_edges = in_sizes[3];
    const int n_inter = in_sizes[4];

    // workspace accumulators
    float* ent_sum = (float*)d_ws;                       // N_ENT*64
    float* ent_cnt = ent_sum + (size_t)N_ENT * DD;       // N_ENT
    float* int_sum = ent_cnt + (size_t)N_ENT;            // N_IT*64
    float* int_cnt = int_sum + (size_t)N_IT * DD;        // N_IT

    // output layout: [final_entity_agg | user_agg | item_kg_agg | item_int_agg]
    float* out_fe = (float*)d_out;
    float* out_ua = out_fe + (size_t)N_ENT * DD;
    float* out_kg = out_ua + (size_t)N_US * DD;
    float* out_in = out_kg + (size_t)N_IT * DD;

    size_t zero_floats = (size_t)N_ENT * DD + N_ENT + (size_t)N_IT * DD + N_IT;
    hipMemsetAsync(d_ws, 0, zero_floats * sizeof(float), stream);
    hipMemsetAsync(out_ua, 0, (size_t)N_US * DD * sizeof(float), stream);

    {
        long long total = (long long)n_edges * 32;
        int blocks = (int)((total + 255) / 256);
        kg_edge_scatter<<<blocks, 256, 0, stream>>>(entity_emb,
                                                    edge_index, edge_index + n_edges,
                                                    edge_type, weight,
                                                    ent_sum, ent_cnt, n_edges);
    }
    {
        long long total = (long long)n_inter * 32;
        int blocks = (int)((total + 255) / 256);
        inter_scatter<<<blocks, 256, 0, stream>>>(user_emb, mat_row, mat_col,
                                                  int_sum, int_cnt, n_inter);
    }
    {
        int blocks = (N_IT + 63) / 64;
        gate_fuse_wmma<<<blocks, 128, 0, stream>>>(ent_sum, ent_cnt, int_sum, int_cnt,
                                                   g1, g2, out_fe, out_kg, out_in, N_IT);
    }
    {
        long long total = (long long)(N_ENT - N_IT) * DD;
        int blocks = (int)((total + 255) / 256);
        att_mean<<<blocks, 256, 0, stream>>>(ent_sum, ent_cnt, out_fe, N_IT, N_ENT);
    }
    {
        long long total = (long long)n_inter * 32;
        int blocks = (int)((total + 255) / 256);
        user_agg_scatter<<<blocks, 256, 0, stream>>>(out_fe, mat_row, mat_col,
                                                     out_ua, n_inter);
    }
}